// PoseEstimator_3204045603271
// MI455X (gfx1250) — compile-verified
//
#include <hip/hip_runtime.h>
#include <math.h>

#define NUM_PARTS  8
#define NUM_KP     32768
#define BETA_DIM   64
#define NUM_JOINTS 7
#define N_FLAT     (NUM_PARTS * NUM_KP * 3)   /* 786432 flat norm_kp elements */

typedef __attribute__((ext_vector_type(2))) float v2f;
typedef __attribute__((ext_vector_type(8))) float v8f;

/* ---- d_out float offsets (reference tuple flattened in return order) ---- */
#define OFF_OBJ        0
#define OFF_BASET      1
#define OFF_REL        17
#define OFF_NEW_ANCHOR 129
#define OFF_NEW_AXIS   150
#define OFF_JLOC       171
#define OFF_JAXIS      192
#define OFF_NORMKP     213

/* ---- workspace float offsets ---- */
#define WS_T       0     /* 8 transforms x 16 floats (T[0]=base_T, T[p]=child_T[p-1]) */
#define WS_REG     128   /* beta regularizer term */
#define WS_PART    256   /* per-block partial sums for the residual reduction */
#define NBLK_RES   1024  /* residual kernel grid size */

/* ------------------------------------------------------------------ */
/* Kernel 1: all the small matrix algebra, single thread (O(3k) FLOPs) */
/* ------------------------------------------------------------------ */
__global__ void k_small(const float* __restrict__ base_r_quat,
                        const float* __restrict__ base_t,
                        const float* __restrict__ joint_state,
                        const float* __restrict__ beta,
                        const float* __restrict__ W_loc,
                        const float* __restrict__ W_axis,
                        float* __restrict__ out,
                        float* __restrict__ ws)
{
    if (threadIdx.x != 0) return;

    /* joint_loc / joint_axis = beta . W  (64 x 7 x 3) */
    float jl[NUM_JOINTS][3], ja[NUM_JOINTS][3];
    for (int j = 0; j < NUM_JOINTS; ++j)
        for (int d = 0; d < 3; ++d) {
            float l = 0.f, a = 0.f;
            for (int e = 0; e < BETA_DIM; ++e) {
                float be = beta[e];
                l += be * W_loc [e * (NUM_JOINTS * 3) + j * 3 + d];
                a += be * W_axis[e * (NUM_JOINTS * 3) + j * 3 + d];
            }
            jl[j][d] = l; ja[j][d] = a;
        }

    float jan[NUM_JOINTS][3];
    for (int j = 0; j < NUM_JOINTS; ++j) {
        float n = sqrtf(ja[j][0]*ja[j][0] + ja[j][1]*ja[j][1] + ja[j][2]*ja[j][2]);
        for (int d = 0; d < 3; ++d) jan[j][d] = ja[j][d] / n;
    }

    /* quaternion -> R */
    float q0 = base_r_quat[0], q1 = base_r_quat[1], q2 = base_r_quat[2], q3 = base_r_quat[3];
    float qn = sqrtf(q0*q0 + q1*q1 + q2*q2 + q3*q3);
    float a = q0/qn, b = q1/qn, c = q2/qn, d = q3/qn;
    float R[3][3] = {
        {1.f - 2.f*(c*c + d*d), 2.f*(b*c - a*d),       2.f*(a*c + b*d)},
        {2.f*(b*c + a*d),       1.f - 2.f*(b*b + d*d), 2.f*(c*d - a*b)},
        {2.f*(b*d - a*c),       2.f*(a*b + c*d),       1.f - 2.f*(b*b + c*c)}};
    float t0 = base_t[0], t1 = base_t[1], t2 = base_t[2];

    float baseT[16] = { R[0][0],R[0][1],R[0][2],t0,
                        R[1][0],R[1][1],R[1][2],t1,
                        R[2][0],R[2][1],R[2][2],t2,
                        0.f,0.f,0.f,1.f };

    for (int i = 0; i < 16; ++i) { out[OFF_BASET + i] = baseT[i]; ws[WS_T + i] = baseT[i]; }

    for (int j = 0; j < NUM_JOINTS; ++j) {
        /* new_anchor = R*loc + t ; new_axis = R*axis_n */
        float aa = R[0][0]*jl[j][0] + R[0][1]*jl[j][1] + R[0][2]*jl[j][2] + t0;
        float bb = R[1][0]*jl[j][0] + R[1][1]*jl[j][1] + R[1][2]*jl[j][2] + t1;
        float cc = R[2][0]*jl[j][0] + R[2][1]*jl[j][1] + R[2][2]*jl[j][2] + t2;
        float u  = R[0][0]*jan[j][0] + R[0][1]*jan[j][1] + R[0][2]*jan[j][2];
        float v  = R[1][0]*jan[j][0] + R[1][1]*jan[j][1] + R[1][2]*jan[j][2];
        float w  = R[2][0]*jan[j][0] + R[2][1]*jan[j][1] + R[2][2]*jan[j][2];

        out[OFF_NEW_ANCHOR + j*3 + 0] = aa; out[OFF_NEW_ANCHOR + j*3 + 1] = bb; out[OFF_NEW_ANCHOR + j*3 + 2] = cc;
        out[OFF_NEW_AXIS   + j*3 + 0] = u;  out[OFF_NEW_AXIS   + j*3 + 1] = v;  out[OFF_NEW_AXIS   + j*3 + 2] = w;
        out[OFF_JLOC  + j*3 + 0] = jl[j][0];  out[OFF_JLOC  + j*3 + 1] = jl[j][1];  out[OFF_JLOC  + j*3 + 2] = jl[j][2];
        out[OFF_JAXIS + j*3 + 0] = jan[j][0]; out[OFF_JAXIS + j*3 + 1] = jan[j][1]; out[OFF_JAXIS + j*3 + 2] = jan[j][2];

        float th = -joint_state[j];
        float cs = cosf(th), sn = sinf(th), omc = 1.f - cs;

        float rel[16];
        rel[0]  = u*u + (v*v + w*w)*cs;
        rel[1]  = u*v*omc - w*sn;
        rel[2]  = u*w*omc + v*sn;
        rel[3]  = (aa*(v*v + w*w) - u*(bb*v + cc*w))*omc + (bb*w - cc*v)*sn;
        rel[4]  = u*v*omc + w*sn;
        rel[5]  = v*v + (u*u + w*w)*cs;
        rel[6]  = v*w*omc - u*sn;
        rel[7]  = (bb*(u*u + w*w) - v*(aa*u + cc*w))*omc + (cc*u - aa*w)*sn;
        rel[8]  = u*w*omc - v*sn;
        rel[9]  = v*w*omc + u*sn;
        rel[10] = w*w + (u*u + v*v)*cs;
        rel[11] = (cc*(u*u + v*v) - w*(aa*u + bb*v))*omc + (aa*v - bb*u)*sn;
        rel[12] = 0.f; rel[13] = 0.f; rel[14] = 0.f; rel[15] = 1.f;

        for (int i = 0; i < 16; ++i) out[OFF_REL + j*16 + i] = rel[i];

        /* child_T = rel @ base_T */
        for (int r = 0; r < 4; ++r)
            for (int cl = 0; cl < 4; ++cl) {
                float s = 0.f;
                for (int kk = 0; kk < 4; ++kk) s += rel[r*4 + kk] * baseT[kk*4 + cl];
                ws[WS_T + (j + 1)*16 + r*4 + cl] = s;
            }
    }

    float sb2 = 0.f;
    for (int e = 0; e < BETA_DIM; ++e) sb2 += beta[e]*beta[e];
    ws[WS_REG] = (sb2 / (float)BETA_DIM) * 0.001f;
}

/* ------------------------------------------------------------------------ */
/* Kernel 2: norm_kp = beta . W_kp  via chained V_WMMA_F32_16X16X4_F32.     */
/* A[m][k] = beta[4i+k] broadcast over rows -> every D row holds the result.*/
/* Per wave: 4 D tiles = 64 outputs; 16 K-steps x 4 tiles = 64 WMMAs.       */
/* One coalesced streaming pass over W_kp (201 MB) -> bandwidth-optimal.    */
/* ------------------------------------------------------------------------ */
__global__ void __launch_bounds__(256)
k_normkp_wmma(const float* __restrict__ W,        /* [64][N_FLAT] row-major */
              const float* __restrict__ beta,
              float* __restrict__ out_normkp)
{
    const int lane = threadIdx.x & 31;
    const int wid  = blockIdx.x * 8 + (threadIdx.x >> 5);
    const int half = lane >> 4;     /* 0: lanes 0-15, 1: lanes 16-31 */
    const int n16  = lane & 15;
    const long base = (long)wid * 64;

    const v2f* bp = (const v2f*)beta;   /* beta pairs, 8B aligned */

    v8f c0 = {}, c1 = {}, c2 = {}, c3 = {};

#pragma unroll
    for (int i = 0; i < 16; ++i) {
        /* A slots: VGPR0 <-> K in {0,2}, VGPR1 <-> K in {1,3} (ISA 16x4 f32 layout);
           lanes 0-15 take K{0,1}, lanes 16-31 take K{2,3}. */
        v2f aop = bp[2*i + half];

        const long r0 = (long)(4*i + 2*half) * N_FLAT;  /* row for slot .x */
        const long r1 = r0 + N_FLAT;                    /* row for slot .y */
        const long col = base + n16;

        v2f b0, b1, b2, b3;
        b0.x = W[r0 + col];      b0.y = W[r1 + col];
        b1.x = W[r0 + col + 16]; b1.y = W[r1 + col + 16];
        b2.x = W[r0 + col + 32]; b2.y = W[r1 + col + 32];
        b3.x = W[r0 + col + 48]; b3.y = W[r1 + col + 48];

        c0 = __builtin_amdgcn_wmma_f32_16x16x4_f32(false, aop, false, b0, (short)0, c0, false, false);
        c1 = __builtin_amdgcn_wmma_f32_16x16x4_f32(false, aop, false, b1, (short)0, c1, false, false);
        c2 = __builtin_amdgcn_wmma_f32_16x16x4_f32(false, aop, false, b2, (short)0, c2, false, false);
        c3 = __builtin_amdgcn_wmma_f32_16x16x4_f32(false, aop, false, b3, (short)0, c3, false, false);
    }

    /* D row 0 (lanes 0-15, VGPR0) carries the result; all rows are equal. */
    if (lane < 16) {
        out_normkp[base + n16]      = c0[0];
        out_normkp[base + 16 + n16] = c1[0];
        out_normkp[base + 32 + n16] = c2[0];
        out_normkp[base + 48 + n16] = c3[0];
    }
}

/* ------------------------------------------------------------------ */
/* Kernel 3: weighted residual norm per keypoint + per-block partials */
/* 128 blocks per part -> part index is block-uniform (scalar T load) */
/* ------------------------------------------------------------------ */
__global__ void __launch_bounds__(256)
k_residual(const float* __restrict__ pred_kp,
           const float* __restrict__ kp_weight,
           const float* __restrict__ normkp,
           const float* __restrict__ ws_T,
           float* __restrict__ partials)
{
    __shared__ float red[256];
    const int p  = blockIdx.x >> 7;                       /* 1024/128 = 8 parts */
    const int k  = ((blockIdx.x & 127) << 8) + threadIdx.x;
    const long pk = (long)p * NUM_KP + k;

    const float* T = ws_T + p * 16;                        /* block-uniform */
    const float* x  = normkp  + pk * 3;
    const float* pr = pred_kp + pk * 3;
    float x0 = x[0], x1 = x[1], x2 = x[2];

    float r0 = T[0]*x0 + T[1]*x1 + T[2] *x2 + T[3]  - pr[0];
    float r1 = T[4]*x0 + T[5]*x1 + T[6] *x2 + T[7]  - pr[1];
    float r2 = T[8]*x0 + T[9]*x1 + T[10]*x2 + T[11] - pr[2];

    red[threadIdx.x] = sqrtf(r0*r0 + r1*r1 + r2*r2) * kp_weight[pk];
    __syncthreads();
    for (int off = 128; off > 0; off >>= 1) {
        if (threadIdx.x < off) red[threadIdx.x] += red[threadIdx.x + off];
        __syncthreads();
    }
    if (threadIdx.x == 0) partials[blockIdx.x] = red[0];
}

/* ------------------------------------------------------- */
/* Kernel 4: deterministic final reduction -> all_obj       */
/* ------------------------------------------------------- */
__global__ void k_final(const float* __restrict__ partials,
                        const float* __restrict__ ws_reg,
                        float* __restrict__ out)
{
    __shared__ float red[256];
    float s = 0.f;
    for (int i = threadIdx.x; i < NBLK_RES; i += 256) s += partials[i];
    red[threadIdx.x] = s;
    __syncthreads();
    for (int off = 128; off > 0; off >>= 1) {
        if (threadIdx.x < off) red[threadIdx.x] += red[threadIdx.x + off];
        __syncthreads();
    }
    if (threadIdx.x == 0)
        out[OFF_OBJ] = red[0] / (float)(NUM_PARTS * NUM_KP) + ws_reg[0];
}

/* ------------------------------------------------------- */
extern "C" void kernel_launch(void* const* d_in, const int* in_sizes, int n_in,
                              void* d_out, int out_size, void* d_ws, size_t ws_size,
                              hipStream_t stream)
{
    const float* pred_kp     = (const float*)d_in[0];
    const float* base_r_quat = (const float*)d_in[1];
    const float* base_t      = (const float*)d_in[2];
    const float* joint_state = (const float*)d_in[3];
    const float* beta        = (const float*)d_in[4];
    const float* kp_weight   = (const float*)d_in[5];
    const float* W_kp        = (const float*)d_in[6];
    const float* W_loc       = (const float*)d_in[7];
    const float* W_axis      = (const float*)d_in[8];

    float* out = (float*)d_out;
    float* ws  = (float*)d_ws;

    k_small<<<1, 32, 0, stream>>>(base_r_quat, base_t, joint_state, beta,
                                  W_loc, W_axis, out, ws);

    /* 1536 blocks * 8 waves * 64 outputs = 786432 = N_FLAT */
    k_normkp_wmma<<<N_FLAT / 512, 256, 0, stream>>>(W_kp, beta, out + OFF_NORMKP);

    k_residual<<<NBLK_RES, 256, 0, stream>>>(pred_kp, kp_weight, out + OFF_NORMKP,
                                             ws + WS_T, ws + WS_PART);

    k_final<<<1, 256, 0, stream>>>(ws + WS_PART, ws + WS_REG, out);
}